// TransformerEncoder_90108413870210
// MI455X (gfx1250) — compile-verified
//
#include <hip/hip_runtime.h>
#include <hip/hip_bf16.h>
#include <math.h>

// Problem sizes (fixed by the reference).
#define B_SZ 1024
#define C_SZ 32
#define K_SZ 16
#define D_SZ 256

typedef __attribute__((ext_vector_type(2))) float v2f;
typedef __attribute__((ext_vector_type(8))) float v8f;

// ---------------------------------------------------------------------------
// Kernel 1: Eq[b][c][d] = sum_e emb[b][e] * W1[c][e][d]      (e in [0,256))
// One wave -> one 16x16 output tile, 64 x V_WMMA_F32_16X16X4_F32 over K=256.
// grid = (B/16, D/(16*8), C), block = 256 (8 waves, each a distinct n-tile).
// ---------------------------------------------------------------------------
__global__ __launch_bounds__(256) void eq_gemm_kernel(const float* __restrict__ emb,
                                                      const float* __restrict__ W1,
                                                      float* __restrict__ Eq) {
    const int lane = threadIdx.x & 31;
    const int wave = threadIdx.x >> 5;
    const int c    = blockIdx.z;
    const int m0   = blockIdx.x << 4;
    const int n0   = ((blockIdx.y << 3) + wave) << 4;
    const int row  = lane & 15;   // M for A-lanes / N for B-lanes
    const int khi  = lane >> 4;   // selects K pair {0,1} vs {2,3}

    // A[m][k]: lane holds (m=row, k=2*khi+{0,1})
    const float* Ap = emb + (size_t)(m0 + row) * D_SZ + (khi << 1);
    // B[k][n]: lane holds (k=2*khi+{0,1}, n=row)
    const float* Bp = W1 + (size_t)c * (2 * D_SZ * D_SZ)
                         + (size_t)(khi << 1) * D_SZ + (n0 + row);

    v8f acc = {};
    #pragma unroll 4
    for (int e0 = 0; e0 < D_SZ; e0 += 4) {
        v2f a, b;
        a.x = Ap[e0];
        a.y = Ap[e0 + 1];
        b.x = Bp[(size_t)e0 * D_SZ];
        b.y = Bp[(size_t)(e0 + 1) * D_SZ];
        acc = __builtin_amdgcn_wmma_f32_16x16x4_f32(
            /*neg_a=*/false, a, /*neg_b=*/false, b,
            /*c_mod=*/(short)0, acc, /*reuse_a=*/false, /*reuse_b=*/false);
    }

    // D layout: VGPR j -> M = j + 8*khi, N = row
    #pragma unroll
    for (int j = 0; j < 8; ++j) {
        const int m = m0 + j + (khi << 3);
        Eq[((size_t)m * C_SZ + c) * D_SZ + n0 + row] = acc[j];
    }
}

// ---------------------------------------------------------------------------
// Kernel 2: Pp[c][k][d] = sum_e proto[c][k][e] * W1[c][256+e][d] + b1[c][d]
// M = K_SZ = 16 exactly one tile.  grid = (D/(16*8), C), block = 256.
// ---------------------------------------------------------------------------
__global__ __launch_bounds__(256) void pp_gemm_kernel(const float* __restrict__ proto,
                                                      const float* __restrict__ W1,
                                                      const float* __restrict__ b1,
                                                      float* __restrict__ Pp) {
    const int lane = threadIdx.x & 31;
    const int wave = threadIdx.x >> 5;
    const int c    = blockIdx.y;
    const int n0   = ((blockIdx.x << 3) + wave) << 4;
    const int row  = lane & 15;
    const int khi  = lane >> 4;

    const float* Ap = proto + (size_t)(c * K_SZ + row) * D_SZ + (khi << 1);
    const float* Bp = W1 + (size_t)c * (2 * D_SZ * D_SZ)
                         + (size_t)(D_SZ + (khi << 1)) * D_SZ + (n0 + row);

    v8f acc = {};
    #pragma unroll 4
    for (int e0 = 0; e0 < D_SZ; e0 += 4) {
        v2f a, b;
        a.x = Ap[e0];
        a.y = Ap[e0 + 1];
        b.x = Bp[(size_t)e0 * D_SZ];
        b.y = Bp[(size_t)(e0 + 1) * D_SZ];
        acc = __builtin_amdgcn_wmma_f32_16x16x4_f32(
            false, a, false, b, (short)0, acc, false, false);
    }

    const float bias = b1[(size_t)c * D_SZ + n0 + row];  // depends on N only
    #pragma unroll
    for (int j = 0; j < 8; ++j) {
        const int k = j + (khi << 3);
        Pp[((size_t)c * K_SZ + k) * D_SZ + n0 + row] = acc[j] + bias;
    }
}

// ---------------------------------------------------------------------------
// Kernel 3: fused tanh-MLP attention + softmax + prototype mix + distance.
// One wave per (b,c); each lane owns 8 contiguous d's (256/32).
// grid = (B, C/8), block = 256 (8 waves -> 8 classes per block).
// ---------------------------------------------------------------------------
__global__ __launch_bounds__(256) void attn_dist_kernel(const float* __restrict__ emb,
                                                        const float* __restrict__ proto,
                                                        const float* __restrict__ w2,
                                                        const float* __restrict__ b2,
                                                        const float* __restrict__ temp,
                                                        const unsigned char* __restrict__ valid,
                                                        const float* __restrict__ Eq,
                                                        const float* __restrict__ Pp,
                                                        float* __restrict__ out) {
    const int lane = threadIdx.x & 31;
    const int wave = threadIdx.x >> 5;
    const int b    = blockIdx.x;
    const int c    = (blockIdx.y << 3) + wave;
    const int d0   = lane << 3;

    float e8[8], q8[8], w8[8];
    #pragma unroll
    for (int i = 0; i < 8; ++i) {
        e8[i] = emb[(size_t)b * D_SZ + d0 + i];
        q8[i] = Eq[((size_t)b * C_SZ + c) * D_SZ + d0 + i];
        w8[i] = w2[(size_t)c * D_SZ + d0 + i];
    }

    // scores[k] = sum_d tanh(Eq + Pp') * w2 + b2
    float sc[K_SZ];
    const float bias2 = b2[c];
    #pragma unroll
    for (int k = 0; k < K_SZ; ++k) {
        const float* pr = Pp + ((size_t)c * K_SZ + k) * D_SZ + d0;
        float part = 0.0f;
        #pragma unroll
        for (int i = 0; i < 8; ++i)
            part += tanhf(q8[i] + pr[i]) * w8[i];
        #pragma unroll
        for (int off = 16; off > 0; off >>= 1)
            part += __shfl_xor(part, off, 32);
        float s = part + bias2;
        if (!valid[c * K_SZ + k]) s = -__builtin_inff();
        sc[k] = s;
    }

    // softmax over K=16 (all lanes hold all 16 scores after the reduction)
    float mx = sc[0];
    #pragma unroll
    for (int k = 1; k < K_SZ; ++k) mx = fmaxf(mx, sc[k]);
    float sum = 0.0f;
    #pragma unroll
    for (int k = 0; k < K_SZ; ++k) { sc[k] = expf(sc[k] - mx); sum += sc[k]; }
    const float inv = 1.0f / sum;

    // final_proto = attn @ prototypes ; squared distance to embedding
    float fp[8] = {0, 0, 0, 0, 0, 0, 0, 0};
    #pragma unroll
    for (int k = 0; k < K_SZ; ++k) {
        const float a = sc[k] * inv;
        const float* pr = proto + ((size_t)c * K_SZ + k) * D_SZ + d0;
        #pragma unroll
        for (int i = 0; i < 8; ++i) fp[i] += a * pr[i];
    }
    float dd = 0.0f;
    #pragma unroll
    for (int i = 0; i < 8; ++i) {
        const float df = e8[i] - fp[i];
        dd += df * df;
    }
    #pragma unroll
    for (int off = 16; off > 0; off >>= 1)
        dd += __shfl_xor(dd, off, 32);

    if (lane == 0)
        out[(size_t)b * C_SZ + c] = -sqrtf(dd) * temp[0];
}

// ---------------------------------------------------------------------------
extern "C" void kernel_launch(void* const* d_in, const int* in_sizes, int n_in,
                              void* d_out, int out_size, void* d_ws, size_t ws_size,
                              hipStream_t stream) {
    (void)in_sizes; (void)n_in; (void)out_size; (void)ws_size;

    const float*         emb   = (const float*)d_in[0];   // [B, D]
    const float*         proto = (const float*)d_in[1];   // [C, K, D]
    const float*         W1    = (const float*)d_in[2];   // [C, 2D, D]
    const float*         b1    = (const float*)d_in[3];   // [C, D]
    const float*         w2    = (const float*)d_in[4];   // [C, D]
    const float*         b2    = (const float*)d_in[5];   // [C]
    const float*         temp  = (const float*)d_in[6];   // [1]
    const unsigned char* valid = (const unsigned char*)d_in[7]; // [C, K] bool
    float*               out   = (float*)d_out;           // [B, C]

    // Workspace: Pp' (C*K*D floats) then Eq (B*C*D floats)  ~34 MB total.
    float* Pp = (float*)d_ws;
    float* Eq = Pp + (size_t)C_SZ * K_SZ * D_SZ;

    // Eq GEMM: grid (B/16, D/128, C), 8 waves per block.
    eq_gemm_kernel<<<dim3(B_SZ / 16, D_SZ / 128, C_SZ), 256, 0, stream>>>(emb, W1, Eq);
    // Pp GEMM (+b1): grid (D/128, C).
    pp_gemm_kernel<<<dim3(D_SZ / 128, C_SZ), 256, 0, stream>>>(proto, W1, b1, Pp);
    // Fused attention/softmax/distance: one wave per (b,c).
    attn_dist_kernel<<<dim3(B_SZ, C_SZ / 8), 256, 0, stream>>>(
        emb, proto, w2, b2, temp, valid, Eq, Pp, out);
}